// LlamaAttention_heavy_hitter_16836271801012
// MI455X (gfx1250) — compile-verified
//
#include <hip/hip_runtime.h>
#include <stdint.h>

// ---------------- problem constants ----------------
#define B_      4
#define H_      4096
#define NH_     32
#define NKV_    8
#define D_      128
#define S_      16384
#define L_      (S_ + 1)          // 16385 (cache + new token)
#define LP_     16416             // padded score-row stride (mult of 32)
#define GRP_    4                 // NH/NKV query heads per kv head
#define HEAVY_  2048
#define INIT_   128
#define LOCAL_  128
#define NEGV    (-3.4028234663852886e38f)
#define CHUNK_  64
#define NCHUNK_ (S_ / CHUNK_ + 1) // 257 (last chunk = new token)

typedef unsigned int uint;
typedef float v2f  __attribute__((ext_vector_type(2)));
typedef float v8f  __attribute__((ext_vector_type(8)));
typedef uint  u32x4 __attribute__((ext_vector_type(4)));
typedef int   i32x4 __attribute__((ext_vector_type(4)));
typedef int   i32x8 __attribute__((ext_vector_type(8)));

// ---------------- helpers ----------------
// pseudo_quantize(BITS=4): q = clip(round((t-mn)*scale),0,15); dq = q/scale+mn.
// q/scale == q*step with step = rng/15 -> one rint + clamp + fma, no divide.
__device__ __forceinline__ float dq4(float t, float mn, float scale, float step) {
  float q = rintf((t - mn) * scale);
  q = fminf(fmaxf(q, 0.0f), 15.0f);
  return fmaf(q, step, mn);
}
__device__ __forceinline__ uint fkey(float f) {      // order-preserving float->uint
  uint u = __float_as_uint(f);
  return u ^ ((u >> 31) ? 0xFFFFFFFFu : 0x80000000u);
}
__device__ __forceinline__ float ifkey(uint k) {
  uint u = (k & 0x80000000u) ? (k ^ 0x80000000u) : ~k;
  return __uint_as_float(u);
}

// Tensor Data Mover: 2D tile (f32) global -> LDS, per CDNA5 ISA D# layout.
__device__ __forceinline__ void tdm_load_2d_f32(uint lds_off, const void* gptr,
                                                uint tile_d0, uint tile_rows,
                                                uint tensor_d0, uint tensor_d1,
                                                uint d0_stride) {
  unsigned long long ga = (unsigned long long)(uintptr_t)gptr;
  u32x4 g0;
  g0.x = 1u;                                   // count=1 (valid), user mode
  g0.y = lds_off;                              // lds_addr (bytes)
  g0.z = (uint)(ga & 0xFFFFFFFFull);           // global_addr[31:0]
  g0.w = (uint)((ga >> 32) & 0x01FFFFFFull) | (2u << 30);  // addr[56:32] | type=2
  i32x8 g1;
  g1[0] = (int)(2u << 16);                     // wg_mask=0, data_size=2 (4B)
  g1[1] = (int)((tensor_d0 & 0xFFFFu) << 16);  // tensor_dim0[15:0] @ bits 48..63
  g1[2] = (int)(((tensor_d0 >> 16) & 0xFFFFu) | ((tensor_d1 & 0xFFFFu) << 16));
  g1[3] = (int)(((tensor_d1 >> 16) & 0xFFFFu) | ((tile_d0 & 0xFFFFu) << 16));
  g1[4] = (int)(tile_rows & 0xFFFFu);          // tile_dim1, tile_dim2=0
  g1[5] = (int)d0_stride;                      // tensor_dim0_stride[31:0]
  g1[6] = 0;
  g1[7] = 0;                                   // dim1 stride unused (2D tile)
  i32x4 g2 = {0, 0, 0, 0};
  i32x4 g3 = {0, 0, 0, 0};
#if defined(__clang_major__) && (__clang_major__ >= 23)
  i32x8 g4 = {0, 0, 0, 0, 0, 0, 0, 0};
  __builtin_amdgcn_tensor_load_to_lds(g0, g1, g2, g3, g4, 0);
#else
  __builtin_amdgcn_tensor_load_to_lds(g0, g1, g2, g3, 0);
#endif
}

// ---------------- 1) fused QKV GEMV (weights read once, all 4 batches) ----------------
__global__ void __launch_bounds__(256) qkv_gemv_kernel(
    const float* __restrict__ hs, const float* __restrict__ Wq,
    const float* __restrict__ Wk, const float* __restrict__ Wv,
    float* __restrict__ qo, float* __restrict__ ko, float* __restrict__ vo) {
  int wid = threadIdx.x >> 5, lane = threadIdx.x & 31;
  int r = blockIdx.x * 8 + wid;                // 0..6143
  const float* W; float* dst; int drow, dstr;
  if (r < 4096)      { W = Wq + (size_t)r * H_;          dst = qo; drow = r;        dstr = NH_ * D_; }
  else if (r < 5120) { W = Wk + (size_t)(r - 4096) * H_; dst = ko; drow = r - 4096; dstr = NKV_ * D_; }
  else               { W = Wv + (size_t)(r - 5120) * H_; dst = vo; drow = r - 5120; dstr = NKV_ * D_; }
  float a0 = 0, a1 = 0, a2 = 0, a3 = 0;
  for (int i = lane; i < H_; i += 32) {
    float w = W[i];
    a0 += w * hs[0 * H_ + i]; a1 += w * hs[1 * H_ + i];
    a2 += w * hs[2 * H_ + i]; a3 += w * hs[3 * H_ + i];
  }
  for (int off = 16; off; off >>= 1) {
    a0 += __shfl_xor(a0, off, 32); a1 += __shfl_xor(a1, off, 32);
    a2 += __shfl_xor(a2, off, 32); a3 += __shfl_xor(a3, off, 32);
  }
  if (lane == 0) {
    dst[0 * dstr + drow] = a0; dst[1 * dstr + drow] = a1;
    dst[2 * dstr + drow] = a2; dst[3 * dstr + drow] = a3;
  }
}

// ---------------- 2) RoPE in place on q and k_new ----------------
__global__ void rope_kernel(float* __restrict__ q, float* __restrict__ k,
                            const float* __restrict__ cosv, const float* __restrict__ sinv) {
  int idx = blockIdx.x * blockDim.x + threadIdx.x;
  if (idx >= B_ * (NH_ + NKV_) * 64) return;
  int d = idx & 63;
  int hk = (idx >> 6) % (NH_ + NKV_);
  int b = idx / (64 * (NH_ + NKV_));
  float* base = (hk < NH_) ? (q + (size_t)b * H_ + hk * D_)
                           : (k + (size_t)b * (NKV_ * D_) + (hk - NH_) * D_);
  float x1 = base[d], x2 = base[d + 64];
  base[d]      = x1 * cosv[d]      - x2 * sinv[d];
  base[d + 64] = x2 * cosv[d + 64] + x1 * sinv[d + 64];
}

// ---------------- 3) scores: TDM K tile -> LDS, 4-bit dequant, f32 WMMA ----------------
// one wave per (b, kv, 16-position tile). D = A(16 pos x 4 d) * B(4 d x 16 heads) + C
// B is the q-group zero-padded to all 16 N columns in LDS -> no divergent DS loads,
// EXEC stays all-ones through the whole WMMA loop.
__global__ void __launch_bounds__(32) scores_kernel(
    const float* __restrict__ kcache, const float* __restrict__ knew,
    const float* __restrict__ qrot, float* __restrict__ scores) {
  __shared__ float smem[16 * D_ + 16 * D_];     // single LDS object -> base offset 0
  float* KT = smem;                              // 16x128 K tile (TDM dest at offset 0)
  float* QG = smem + 16 * D_;                    // 16x128 q (rows 0-3 real, 4-15 zero)
  const int lane = threadIdx.x;
  const int pos0 = blockIdx.x * 16;
  const int kv = blockIdx.y, b = blockIdx.z;

  if (pos0 < S_) {                               // async DMA of the K tile
    const float* gp = kcache + (((size_t)b * NKV_ + kv) * S_ + pos0) * (size_t)D_;
    tdm_load_2d_f32(0u, gp, D_, 16, D_, (uint)(S_ - pos0), D_);
  }
  for (int t = lane; t < 16 * D_; t += 32)       // stage q group while DMA runs
    QG[t] = (t < GRP_ * D_) ? qrot[(size_t)b * H_ + (kv * GRP_) * D_ + t] : 0.0f;
  if (pos0 < S_) {
    __builtin_amdgcn_s_wait_tensorcnt(0);
  } else {                                       // last tile: the new RoPE'd key
    for (int t = lane; t < 16 * D_; t += 32)
      KT[t] = (t < D_) ? knew[(size_t)b * (NKV_ * D_) + kv * D_ + t] : 0.0f;
  }
  __syncthreads();

  // per-row min/max over D=128: lane pair (l, l^16) each scans half a row
  const int row = lane & 15, half = lane >> 4;
  float mn = 3.4e38f, mx = -3.4e38f;
  const float* rp = KT + row * D_ + half * 64;
  #pragma unroll 8
  for (int i = 0; i < 64; ++i) { float v = rp[i]; mn = fminf(mn, v); mx = fmaxf(mx, v); }
  mn = fminf(mn, __shfl_xor(mn, 16, 32));
  mx = fmaxf(mx, __shfl_xor(mx, 16, 32));
  float rng = mx - mn; if (rng == 0.0f) rng = 1.0f;
  const float scl = 15.0f / rng;                 // quant scale
  const float stp = rng * (1.0f / 15.0f);        // dequant step (q/scale == q*step)

  const int n = lane & 15, ksub = lane >> 4;     // B frag: N=head, K rows {ksub, 2+ksub}
  v8f acc = {0, 0, 0, 0, 0, 0, 0, 0};
  for (int d0 = 0; d0 < D_; d0 += 4) {
    v2f a, bb;
    a.x = dq4(KT[row * D_ + d0 + ksub * 2 + 0], mn, scl, stp);  // A frag K={0,1}/{2,3}
    a.y = dq4(KT[row * D_ + d0 + ksub * 2 + 1], mn, scl, stp);
    bb.x = QG[n * D_ + d0 + ksub];
    bb.y = QG[n * D_ + d0 + 2 + ksub];
    acc = __builtin_amdgcn_wmma_f32_16x16x4_f32(false, a, false, bb,
                                                (short)0, acc, false, false);
  }
  const float sinv = 0.08838834764831845f;       // 1/sqrt(128)
  if (n < GRP_) {                                // D frag: VGPR r = M=r (+8 for hi lanes)
    const int head = kv * GRP_ + n;
    const size_t rowoff = ((size_t)b * NH_ + head) * LP_;
    const int mbase = (lane >> 4) * 8;
    #pragma unroll
    for (int r = 0; r < 8; ++r) {
      int pos = pos0 + mbase + r;
      if (pos < L_) scores[rowoff + pos] = acc[r] * sinv;
    }
  }
}

// ---------------- 4) exact top-k threshold (radix select) + max + sumexp ----------------
__global__ void __launch_bounds__(256) select_kernel(
    const float* __restrict__ scores, float* __restrict__ stats) {
  __shared__ uint hist[256];
  __shared__ uint aux[2];
  __shared__ float red[256];
  const int tid = threadIdx.x;
  const int bh = blockIdx.x;                     // b*NH + h
  const float* rowp = scores + (size_t)bh * LP_;
  uint prefix = 0, need = HEAVY_;
  for (int pass = 0; pass < 4; ++pass) {
    const int shift = 24 - 8 * pass;
    hist[tid] = 0;
    __syncthreads();
    for (int i = tid; i < L_; i += 256) {
      float s = rowp[i];
      bool alw = (i < INIT_) || (i >= L_ - LOCAL_);
      uint key = fkey(alw ? NEGV : s);           // top_k runs on masked scores
      bool cand = (pass == 0) || (((key ^ prefix) >> (shift + 8)) == 0);
      if (cand) atomicAdd(&hist[(key >> shift) & 255u], 1u);
    }
    __syncthreads();
    if (tid == 0) {
      uint running = 0, chosen = 0, nd = need;
      for (int d = 255; d >= 0; --d) {
        uint c = hist[d];
        if (running + c >= nd) { chosen = (uint)d; nd -= running; break; }
        running += c;
      }
      aux[0] = prefix | (chosen << shift);
      aux[1] = nd;
    }
    __syncthreads();
    prefix = aux[0]; need = aux[1];
    __syncthreads();
  }
  const float thresh = ifkey(prefix);            // exact 2048th-largest masked score
  float lm = -3.4e38f;
  for (int i = tid; i < L_; i += 256) {
    float s = rowp[i];
    bool keep = (i < INIT_) || (i >= L_ - LOCAL_) || (s >= thresh);
    if (keep) lm = fmaxf(lm, s);
  }
  red[tid] = lm; __syncthreads();
  for (int off = 128; off; off >>= 1) { if (tid < off) red[tid] = fmaxf(red[tid], red[tid + off]); __syncthreads(); }
  const float mx = red[0]; __syncthreads();
  float ls = 0.0f;
  for (int i = tid; i < L_; i += 256) {
    float s = rowp[i];
    bool keep = (i < INIT_) || (i >= L_ - LOCAL_) || (s >= thresh);
    if (keep) ls += expf(s - mx);
  }
  red[tid] = ls; __syncthreads();
  for (int off = 128; off; off >>= 1) { if (tid < off) red[tid] += red[tid + off]; __syncthreads(); }
  if (tid == 0) { stats[bh * 4 + 0] = thresh; stats[bh * 4 + 1] = mx; stats[bh * 4 + 2] = red[0]; }
}

// ---------------- 5) AV: TDM V tile -> LDS, dequant, weighted accumulate ----------------
__global__ void __launch_bounds__(128) av_kernel(
    const float* __restrict__ vcache, const float* __restrict__ vnew,
    const float* __restrict__ scores, const float* __restrict__ stats,
    float* __restrict__ part) {
  __shared__ float smem[CHUNK_ * D_ + 3 * CHUNK_ + GRP_ * CHUNK_];
  float* VT  = smem;                             // 64x128 V tile (LDS offset 0 for TDM)
  float* MN  = smem + CHUNK_ * D_;
  float* SCL = MN + CHUNK_;
  float* STP = SCL + CHUNK_;
  float* WL  = STP + CHUNK_;
  const int tid = threadIdx.x;
  const int c = blockIdx.x, kv = blockIdx.y, b = blockIdx.z;
  const int pos0 = c * CHUNK_;

  if (tid < 32 && pos0 < S_) {                   // wave 0 drives the tensor DMA
    const float* gp = vcache + (((size_t)b * NKV_ + kv) * S_ + pos0) * (size_t)D_;
    tdm_load_2d_f32(0u, gp, D_, CHUNK_, D_, (uint)(S_ - pos0), D_);
    __builtin_amdgcn_s_wait_tensorcnt(0);
  }
  if (pos0 >= S_) {                              // last chunk: the new value row
    for (int t = tid; t < CHUNK_ * D_; t += 128)
      VT[t] = (t < D_) ? vnew[(size_t)b * (NKV_ * D_) + kv * D_ + t] : 0.0f;
  }
  __syncthreads();

  const int wid = tid >> 5, lane = tid & 31;     // per-row quant params, wave per 16 rows
  for (int rr = 0; rr < 16; ++rr) {
    int row = wid * 16 + rr;
    const float* rp = VT + row * D_ + lane * 4;
    float mn = fminf(fminf(rp[0], rp[1]), fminf(rp[2], rp[3]));
    float mx = fmaxf(fmaxf(rp[0], rp[1]), fmaxf(rp[2], rp[3]));
    for (int off = 16; off; off >>= 1) {
      mn = fminf(mn, __shfl_xor(mn, off, 32));
      mx = fmaxf(mx, __shfl_xor(mx, off, 32));
    }
    if (lane == 0) {
      float rng = mx - mn; if (rng == 0.0f) rng = 1.0f;
      MN[row] = mn; SCL[row] = 15.0f / rng; STP[row] = rng * (1.0f / 15.0f);
    }
  }
  for (int idx = tid; idx < GRP_ * CHUNK_; idx += 128) {   // softmax weights (unnormalized)
    int h = idx >> 6, r = idx & (CHUNK_ - 1);
    int pos = pos0 + r;
    float w = 0.0f;
    if (pos < L_) {
      int bh = b * NH_ + kv * GRP_ + h;
      float s  = scores[(size_t)bh * LP_ + pos];
      float th = stats[bh * 4 + 0], mx = stats[bh * 4 + 1];
      bool keep = (pos < INIT_) || (pos >= L_ - LOCAL_) || (s >= th);
      if (keep) w = expf(s - mx);
    }
    WL[idx] = w;
  }
  __syncthreads();

  const int d = tid;                             // 128 threads <-> 128 dims
  float a0 = 0, a1 = 0, a2 = 0, a3 = 0;
  for (int r = 0; r < CHUNK_; ++r) {
    float vq = dq4(VT[r * D_ + d], MN[r], SCL[r], STP[r]);
    a0 += WL[0 * CHUNK_ + r] * vq; a1 += WL[1 * CHUNK_ + r] * vq;
    a2 += WL[2 * CHUNK_ + r] * vq; a3 += WL[3 * CHUNK_ + r] * vq;
  }
  size_t pbase = (((size_t)b * NH_ + kv * GRP_) * NCHUNK_ + c) * D_ + d;
  const size_t hstride = (size_t)NCHUNK_ * D_;
  part[pbase + 0 * hstride] = a0; part[pbase + 1 * hstride] = a1;
  part[pbase + 2 * hstride] = a2; part[pbase + 3 * hstride] = a3;
}

// ---------------- 6) deterministic partial reduction + softmax normalize ----------------
__global__ void __launch_bounds__(128) reduce_attn_kernel(
    const float* __restrict__ part, const float* __restrict__ stats,
    float* __restrict__ attn) {
  const int bh = blockIdx.x, d = threadIdx.x;
  float s = 0.0f;
  const float* p = part + (size_t)bh * NCHUNK_ * D_ + d;
  for (int c = 0; c < NCHUNK_; ++c) s += p[(size_t)c * D_];
  attn[(size_t)bh * D_ + d] = s / stats[bh * 4 + 2];
}

// ---------------- 7) output projection (Wo read once, all 4 batches) ----------------
__global__ void __launch_bounds__(256) outproj_kernel(
    const float* __restrict__ attn, const float* __restrict__ Wo,
    float* __restrict__ out) {
  const int wid = threadIdx.x >> 5, lane = threadIdx.x & 31;
  const int j = blockIdx.x * 8 + wid;
  const float* wr = Wo + (size_t)j * H_;
  float a0 = 0, a1 = 0, a2 = 0, a3 = 0;
  for (int i = lane; i < H_; i += 32) {
    __builtin_prefetch(wr + i + 2048, 0, 1);     // global_prefetch_b8 path
    float w = wr[i];
    a0 += w * attn[0 * H_ + i]; a1 += w * attn[1 * H_ + i];
    a2 += w * attn[2 * H_ + i]; a3 += w * attn[3 * H_ + i];
  }
  for (int off = 16; off; off >>= 1) {
    a0 += __shfl_xor(a0, off, 32); a1 += __shfl_xor(a1, off, 32);
    a2 += __shfl_xor(a2, off, 32); a3 += __shfl_xor(a3, off, 32);
  }
  if (lane == 0) {
    out[0 * H_ + j] = a0; out[1 * H_ + j] = a1;
    out[2 * H_ + j] = a2; out[3 * H_ + j] = a3;
  }
}

// ---------------- host launch ----------------
extern "C" void kernel_launch(void* const* d_in, const int* in_sizes, int n_in,
                              void* d_out, int out_size, void* d_ws, size_t ws_size,
                              hipStream_t stream) {
  (void)in_sizes; (void)n_in; (void)out_size; (void)ws_size;
  const float* hs     = (const float*)d_in[0];
  const float* kcache = (const float*)d_in[1];
  const float* vcache = (const float*)d_in[2];
  const float* cosv   = (const float*)d_in[3];
  const float* sinv   = (const float*)d_in[4];
  const float* Wq     = (const float*)d_in[5];
  const float* Wk     = (const float*)d_in[6];
  const float* Wv     = (const float*)d_in[7];
  const float* Wo     = (const float*)d_in[8];

  float* ws     = (float*)d_ws;
  float* q      = ws;                                      // B*NH*D    = 16384
  float* k      = q + (size_t)B_ * NH_ * D_;               // B*NKV*D   = 4096
  float* v      = k + (size_t)B_ * NKV_ * D_;              // B*NKV*D   = 4096
  float* scores = v + (size_t)B_ * NKV_ * D_;              // B*NH*LP   = 2101248
  float* stats  = scores + (size_t)B_ * NH_ * LP_;         // B*NH*4    = 512
  float* part   = stats + (size_t)B_ * NH_ * 4;            // B*NH*NCHUNK*D = 4210688
  float* attn   = part + (size_t)B_ * NH_ * NCHUNK_ * D_;  // B*NH*D    = 16384
  float* out    = (float*)d_out;

  qkv_gemv_kernel<<<dim3((H_ + 2 * NKV_ * D_) / 8), dim3(256), 0, stream>>>(
      hs, Wq, Wk, Wv, q, k, v);
  rope_kernel<<<dim3((B_ * (NH_ + NKV_) * 64 + 255) / 256), dim3(256), 0, stream>>>(
      q, k, cosv, sinv);
  scores_kernel<<<dim3((L_ + 15) / 16, NKV_, B_), dim3(32), 0, stream>>>(
      kcache, k, q, scores);
  select_kernel<<<dim3(B_ * NH_), dim3(256), 0, stream>>>(scores, stats);
  av_kernel<<<dim3(NCHUNK_, NKV_, B_), dim3(128), 0, stream>>>(
      vcache, v, scores, stats, part);
  reduce_attn_kernel<<<dim3(B_ * NH_), dim3(D_), 0, stream>>>(part, stats, attn);
  outproj_kernel<<<dim3(H_ / 8), dim3(256), 0, stream>>>(attn, Wo, out);
}